// RESKnorm_40956808135039
// MI455X (gfx1250) — compile-verified
//
#include <hip/hip_runtime.h>
#include <hip/hip_bf16.h>

typedef __attribute__((ext_vector_type(2))) float v2f;
typedef __attribute__((ext_vector_type(8))) float v8f;

static __host__ __device__ inline int cdiv(int a, int b) { return (a + b - 1) / b; }

// ---------------------------------------------------------------------------
// GEMM: S[N,M] = X[N,K] @ W[K,M] + bias.  One wave -> 16 rows x (NT*16) cols.
// V_WMMA_F32_16X16X4_F32 over K in steps of 4.  W staged in LDS as float2
// K-pairs with padded stride so B fragments are conflict-free ds_load_b64.
// K, M compile-time -> no exec-masked loads in the hot loop (EXEC all-1s).
// Store epilogue: wave-uniform fast path (unmasked) for full row tiles.
// ---------------------------------------------------------------------------
template <int K, int M, int NT>
__global__ __launch_bounds__(256) void gemm_wmma(
    const float* __restrict__ X, const float* __restrict__ W,
    const float* __restrict__ bias, float* __restrict__ S, int N) {
  constexpr int PM2 = M + 16;          // float2 row stride (bank-shift = 32 lanes)
  __shared__ v2f Wl[(K / 2) * PM2];    // {W[2k][c], W[2k+1][c]}

  // ---- cooperative stage of W into LDS (all threads, before any exit) ----
  for (int i = threadIdx.x; i < (K / 2) * M; i += 256) {
    const int kp = i / M;
    const int col = i % M;
    v2f w;
    w.x = W[(size_t)(2 * kp + 0) * M + col];
    w.y = W[(size_t)(2 * kp + 1) * M + col];
    Wl[kp * PM2 + col] = w;
  }
  __syncthreads();

  const int wave = (blockIdx.x * blockDim.x + threadIdx.x) >> 5;
  const int lane = threadIdx.x & 31;
  const int row0 = wave * 16;
  if (row0 >= N) return;  // uniform per wave -> EXEC all-1s at every WMMA

  const int half = lane >> 4;   // 0 or 1
  const int lrow = lane & 15;   // 0..15

  v8f acc[NT];
#pragma unroll
  for (int t = 0; t < NT; ++t) acc[t] = (v8f)(0.0f);

  // clamp OOB rows (results masked at store)
  int arow = row0 + lrow;
  if (arow >= N) arow = N - 1;
  const float* __restrict__ xrow = X + (size_t)arow * K + 2 * half;

  // clamp OOB cols (garbage cols never stored) -> unmasked B loads
  int colc[NT];
#pragma unroll
  for (int t = 0; t < NT; ++t) {
    int c = t * 16 + lrow;
    colc[t] = (c < M) ? c : (M - 1);
  }

#pragma unroll 4
  for (int k0 = 0; k0 < K; k0 += 4) {
    v2f a = *(const v2f*)(xrow + k0);            // global_load_b64
    const int kp = (k0 >> 1) + half;             // K-pair index for this half
#pragma unroll
    for (int t = 0; t < NT; ++t) {
      v2f b = Wl[kp * PM2 + colc[t]];            // ds_load_2addr_b64
      acc[t] = __builtin_amdgcn_wmma_f32_16x16x4_f32(
          /*neg_a=*/false, a, /*neg_b=*/false, b,
          /*c_mod=*/(short)0, acc[t], /*reuse_a=*/false, /*reuse_b=*/false);
    }
  }

  // C/D layout: VGPR v holds row = row0 + v + 8*half, col = t*16 + lane%16
  if (row0 + 16 <= N) {
    // fast path (all waves except the last row tile): unmasked stores,
    // base pointer + compile-time immediate offsets v*M*4
#pragma unroll
    for (int t = 0; t < NT; ++t) {
      const int col = t * 16 + lrow;
      if (col < M) {                       // compile-time true for M%16==0
        const float bv = bias[col];
        float* __restrict__ sp = S + (size_t)(row0 + 8 * half) * M + col;
#pragma unroll
        for (int v = 0; v < 8; ++v) {
          sp[v * M] = acc[t][v] + bv;      // global_store_b32, imm offset
        }
      }
    }
  } else {
    // slow path: single trailing wave, per-row masked stores
#pragma unroll
    for (int t = 0; t < NT; ++t) {
      const int col = t * 16 + lrow;
      if (col < M) {
        const float bv = bias[col];
#pragma unroll
        for (int v = 0; v < 8; ++v) {
          const int r = row0 + v + 8 * half;
          if (r < N) S[(size_t)r * M + col] = acc[t][v] + bv;
        }
      }
    }
  }
}

// ---------------------------------------------------------------------------
// Zero-fill
// ---------------------------------------------------------------------------
__global__ void fill_zero(float* __restrict__ p, int n) {
  int i = blockIdx.x * blockDim.x + threadIdx.x;
  if (i < n) p[i] = 0.0f;
}

// ---------------------------------------------------------------------------
// Edge gather-scale-scatter: out[tgt[e]] += mval[e] * S[src[e]]
// One wave per edge; lane handles channels (2*lane, 2*lane+1).
// ---------------------------------------------------------------------------
__global__ __launch_bounds__(256) void edge_scatter(
    const float* __restrict__ S, const int* __restrict__ src,
    const int* __restrict__ tgt, const float* __restrict__ mv,
    float* __restrict__ out, int E, int C) {
  const int e = (blockIdx.x * blockDim.x + threadIdx.x) >> 5;
  const int lane = threadIdx.x & 31;
  if (e >= E) return;
  const int c = lane * 2;
  if (c >= C) return;
  const int s = src[e];
  const int t = tgt[e];
  const float m = mv[e];
  const float* __restrict__ sp = S + (size_t)s * C + c;
  float* op = out + (size_t)t * C + c;
  const float x0 = sp[0] * m;
  const float x1 = sp[1] * m;
#if defined(__HIP_DEVICE_COMPILE__)
  unsafeAtomicAdd(op + 0, x0);   // hardware global_atomic_add_f32
  unsafeAtomicAdd(op + 1, x1);
#else
  atomicAdd(op + 0, x0);
  atomicAdd(op + 1, x1);
#endif
}

// ---------------------------------------------------------------------------
// h = relu(agg)
// ---------------------------------------------------------------------------
__global__ void relu_kernel(const float* __restrict__ agg, float* __restrict__ h, int n) {
  int i = blockIdx.x * blockDim.x + threadIdx.x;
  if (i < n) h[i] = fmaxf(agg[i], 0.0f);
}

// ---------------------------------------------------------------------------
// out = groupnorm(relu(agg)) * gamma + beta + r    (groups of 2 channels, C=64)
// One thread per (node, group).
// ---------------------------------------------------------------------------
__global__ void gnres_kernel(const float* __restrict__ agg, const float* __restrict__ r,
                             const float* __restrict__ gamma, const float* __restrict__ beta,
                             float* __restrict__ out, int N) {
  int gid = blockIdx.x * blockDim.x + threadIdx.x;
  if (gid >= N * 32) return;
  const int n = gid >> 5;
  const int c = (gid & 31) * 2;
  const size_t base = (size_t)n * 64 + c;
  const float a0 = fmaxf(agg[base + 0], 0.0f);
  const float a1 = fmaxf(agg[base + 1], 0.0f);
  const float mu = 0.5f * (a0 + a1);
  const float d0 = a0 - mu, d1 = a1 - mu;
  const float var = 0.5f * (d0 * d0 + d1 * d1);
  const float inv = rsqrtf(var + 1e-5f);
  out[base + 0] = d0 * inv * gamma[c + 0] + beta[c + 0] + r[base + 0];
  out[base + 1] = d1 * inv * gamma[c + 1] + beta[c + 1] + r[base + 1];
}

// ---------------------------------------------------------------------------
// log_softmax over C classes, one thread per node.
// ---------------------------------------------------------------------------
__global__ void lsm_kernel(const float* __restrict__ agg, float* __restrict__ out,
                           int N, int C) {
  int n = blockIdx.x * blockDim.x + threadIdx.x;
  if (n >= N) return;
  const float* __restrict__ row = agg + (size_t)n * C;
  float mx = -3.402823466e+38f;
  for (int c = 0; c < C; ++c) mx = fmaxf(mx, row[c]);
  float s = 0.0f;
  for (int c = 0; c < C; ++c) s += __expf(row[c] - mx);
  const float l = __logf(s);
  float* __restrict__ orow = out + (size_t)n * C;
  for (int c = 0; c < C; ++c) orow[c] = row[c] - mx - l;
}

// ---------------------------------------------------------------------------
// Host orchestration
// ---------------------------------------------------------------------------
extern "C" void kernel_launch(void* const* d_in, const int* in_sizes, int n_in,
                              void* d_out, int out_size, void* d_ws, size_t ws_size,
                              hipStream_t stream) {
  const int NFEAT = 128, NHID = 64, NCLASS = 40;
  const int N = in_sizes[0] / NFEAT;
  const int E = in_sizes[1];

  const float* x    = (const float*)d_in[0];
  const int*   src  = (const int*)d_in[1];
  const int*   tgt  = (const int*)d_in[2];
  const float* mv   = (const float*)d_in[3];
  const float* W0   = (const float*)d_in[4];
  const float* b0   = (const float*)d_in[5];
  const float* W1   = (const float*)d_in[6];
  const float* b1   = (const float*)d_in[7];
  const float* W2   = (const float*)d_in[8];
  const float* b2   = (const float*)d_in[9];
  const float* W3   = (const float*)d_in[10];
  const float* b3   = (const float*)d_in[11];
  const float* g1   = (const float*)d_in[12];
  const float* be1  = (const float*)d_in[13];
  const float* g2   = (const float*)d_in[14];
  const float* be2  = (const float*)d_in[15];
  float* out = (float*)d_out;

  const size_t NH = (size_t)N * NHID;
  float* hA  = (float*)d_ws;       // [N,64]
  float* hB  = hA + NH;            // [N,64]
  float* sbf = hB + NH;            // [N,64] gemm output
  float* agg = sbf + NH;           // [N,64] scatter accumulator

  const int rowWaves   = cdiv(N, 16);          // waves per gemm
  const int gemmBlocks = cdiv(rowWaves, 8);    // 8 waves / 256-thread block
  const int scatBlocks = cdiv(E, 8);           // 8 edges (waves) per block
  const int zero64     = cdiv((int)NH, 256);
  const int gnBlocks   = cdiv(N * 32, 256);

  // ---- layer 0: h = relu(gcn(x, W0)) ----
  gemm_wmma<128, 64, 4><<<gemmBlocks, 256, 0, stream>>>(x, W0, b0, sbf, N);
  fill_zero<<<zero64, 256, 0, stream>>>(agg, (int)NH);
  edge_scatter<<<scatBlocks, 256, 0, stream>>>(sbf, src, tgt, mv, agg, E, NHID);
  relu_kernel<<<zero64, 256, 0, stream>>>(agg, hA, (int)NH);

  // ---- layer 1: hB = gn(relu(gcn(hA, W1))) + hA ----
  gemm_wmma<64, 64, 4><<<gemmBlocks, 256, 0, stream>>>(hA, W1, b1, sbf, N);
  fill_zero<<<zero64, 256, 0, stream>>>(agg, (int)NH);
  edge_scatter<<<scatBlocks, 256, 0, stream>>>(sbf, src, tgt, mv, agg, E, NHID);
  gnres_kernel<<<gnBlocks, 256, 0, stream>>>(agg, hA, g1, be1, hB, N);

  // ---- layer 2: hA = gn(relu(gcn(hB, W2))) + hB ----
  gemm_wmma<64, 64, 4><<<gemmBlocks, 256, 0, stream>>>(hB, W2, b2, sbf, N);
  fill_zero<<<zero64, 256, 0, stream>>>(agg, (int)NH);
  edge_scatter<<<scatBlocks, 256, 0, stream>>>(sbf, src, tgt, mv, agg, E, NHID);
  gnres_kernel<<<gnBlocks, 256, 0, stream>>>(agg, hB, g2, be2, hA, N);

  // ---- output layer: log_softmax(gcn(hA, W3)) ----
  gemm_wmma<64, 40, 3><<<gemmBlocks, 256, 0, stream>>>(hA, W3, b3, sbf, N);
  const int nOut = N * NCLASS;
  fill_zero<<<cdiv(nOut, 256), 256, 0, stream>>>(agg, nOut);
  edge_scatter<<<scatBlocks, 256, 0, stream>>>(sbf, src, tgt, mv, agg, E, NCLASS);
  lsm_kernel<<<cdiv(N, 256), 256, 0, stream>>>(agg, out, N, NCLASS);
}